// PatchedVisionExpertMLP_65317862638204
// MI455X (gfx1250) — compile-verified
//
#include <hip/hip_runtime.h>

// Problem constants (from reference): B=2, L=2048, D=1024, FF=4096
#define DIM_D   1024
#define DIM_FF  4096
#define SEQ_L   2048
#define N_TOK   4096      // B*L
#define TOK     32        // tokens per block (two 16-row WMMA sub-tiles)
#define FC      1024      // FF chunk held in LDS
#define NCHUNK  (DIM_FF / FC)
#define XS      1028      // LDS row stride (floats): pad 4 -> conflict-free b64 reads
#define HS      1028
#define NTHREADS 512      // 16 waves (wave32)

typedef __attribute__((ext_vector_type(2))) float v2f;
typedef __attribute__((ext_vector_type(8))) float v8f;

// D(16x16,f32) = A(16x4,f32) * B(4x16,f32) + C  -- native fp32 matrix path on CDNA5
__device__ __forceinline__ v8f wmma_f32(v2f a, v2f b, v8f c) {
  return __builtin_amdgcn_wmma_f32_16x16x4_f32(false, a, false, b, (short)0, c,
                                               false, false);
}

// silu(g)*u with hardware transcendentals (v_exp_f32 + v_rcp_f32, ~1 ulp)
__device__ __forceinline__ float silu_mul(float g, float u) {
  float s = __builtin_amdgcn_rcpf(1.0f + __expf(-g));
  return g * s * u;
}

// ---------------------------------------------------------------------------
// Kernel 1: vision mask + stable compaction (single block, LDS prefix scan,
// no atomics => deterministic lists). vis/lang hold flat token indices.
// ---------------------------------------------------------------------------
__global__ __launch_bounds__(1024)
void mask_compact(const int* __restrict__ tt, int* __restrict__ vis,
                  int* __restrict__ lang, int* __restrict__ counts) {
  __shared__ int s[1024];
  const int tid = threadIdx.x;

  int m[4];
  int cnt = 0;
#pragma unroll
  for (int j = 0; j < 4; ++j) {
    int i = tid * 4 + j;
    int l = i & (SEQ_L - 1);
    // vision iff this token AND next token are type 1; last column false
    int v = (tt[i] == 1) && (l + 1 < SEQ_L) && (tt[i + 1] == 1);
    m[j] = v;
    cnt += v;
  }

  s[tid] = cnt;
  __syncthreads();
  for (int off = 1; off < 1024; off <<= 1) {
    int add = (tid >= off) ? s[tid - off] : 0;
    __syncthreads();
    s[tid] += add;
    __syncthreads();
  }
  const int incl = s[tid];
  const int nvis = s[1023];
  int ev = incl - cnt;       // exclusive vision prefix
  int el = tid * 4 - ev;     // exclusive language prefix

#pragma unroll
  for (int j = 0; j < 4; ++j) {
    int i = tid * 4 + j;
    if (m[j]) vis[ev++] = i;
    else      lang[el++] = i;
  }
  if (tid == 0) {
    counts[0] = nvis;
    counts[1] = N_TOK - nvis;
  }
}

// ---------------------------------------------------------------------------
// Kernel 2: fused SwiGLU MLP for one expert over 32 gathered tokens.
//   LDS: Xs[32][1028] gathered activations, Hs[32][1028] FF chunk of
//   silu(gate)*up. FF processed in 4 chunks; down-proj accumulated in
//   64 persistent VGPRs/lane. grid = (N_TOK/TOK, 2 experts).
// ---------------------------------------------------------------------------
__global__ __launch_bounds__(NTHREADS)
void fused_mlp(const float* __restrict__ hs,
               const float* __restrict__ vgate, const float* __restrict__ vup,
               const float* __restrict__ vdown,
               const float* __restrict__ lgate, const float* __restrict__ lup,
               const float* __restrict__ ldown,
               const int* __restrict__ vis, const int* __restrict__ lang,
               const int* __restrict__ counts,
               float* __restrict__ out) {
  const int e = blockIdx.y;
  const int count = counts[e];
  const int t0 = blockIdx.x * TOK;
  if (t0 >= count) return;                    // block-uniform exit

  const int* list = e ? lang : vis;
  const float* wg = e ? lgate : vgate;
  const float* wu = e ? lup   : vup;
  const float* wd = e ? ldown : vdown;

  extern __shared__ float smem[];
  float* Xs = smem;                 // [TOK][XS]
  float* Hs = smem + TOK * XS;      // [TOK][HS]
  __shared__ int tokIdx[TOK];       // gathered token indices (-1 = pad row)

  const int lane  = threadIdx.x & 31;
  const int wave  = threadIdx.x >> 5;
  const int mrow  = lane & 15;                // M / N index within fragment
  const int khalf = (lane & 16) ? 2 : 0;      // K sub-column for A/B frags
  const int half8 = (lane & 16) ? 8 : 0;      // C/D row offset for lanes 16-31
  const int r     = wave & 1;                 // 16-token sub-tile owned by wave
  const int wct   = wave >> 1;                // 0..7 column-task base

  // ---- Phase 0: async-gather 32 token rows of X straight into LDS ----
  // GLOBAL_LOAD_ASYNC_TO_LDS_B128: copies 16B from global vaddr to LDS at vdst, tracked by ASYNCcnt.
  // LDS byte offset = low 32 bits of the flat shared-memory address.
  {
    const int j  = threadIdx.x >> 4;          // 0..31 : token row
    const int lr = threadIdx.x & 15;          // 16 lanes x 16B = 256B per step
    const int p  = t0 + j;
    const bool valid = (p < count);
    const int src = valid ? list[p] : list[t0];   // pad rows = dup row
    if (lr == 0) tokIdx[j] = valid ? src : -1;
    const unsigned long long gbase =
        (unsigned long long)(const void*)(hs + (size_t)src * DIM_D) + lr * 16u;
    const unsigned lbase = (unsigned)(size_t)(void*)(Xs + j * XS) + lr * 16u;
#pragma unroll
    for (int it = 0; it < 16; ++it) {
      unsigned long long ga = gbase + (unsigned)(it * 256);
      unsigned la = lbase + (unsigned)(it * 256);
      asm volatile("global_load_async_to_lds_b128 %0, %1, off"
                   :: "v"(la), "v"(ga) : "memory");
    }
    asm volatile("s_wait_asynccnt 0" ::: "memory");
  }

  // persistent down-proj accumulators: 8 tasks x (16x16 f32 tile)
  v8f accd[8];
  const v8f vz = {0.f, 0.f, 0.f, 0.f, 0.f, 0.f, 0.f, 0.f};
#pragma unroll
  for (int i = 0; i < 8; ++i) accd[i] = vz;

  for (int c = 0; c < NCHUNK; ++c) {
    __syncthreads();   // Xs/tokIdx ready (c==0); Hs free to overwrite (c>0)

    // ---- Phase A: Hs = silu(X*Wg^T) * (X*Wu^T) for FF cols [c*FC, c*FC+FC)
    for (int i = 0; i < 8; ++i) {
      const int ct = wct + 8 * i;             // 0..63 : 16-wide FF col tile
      const int fglob = c * FC + ct * 16 + mrow;
      const float* pg = wg + (size_t)fglob * DIM_D + khalf;
      const float* pu = wu + (size_t)fglob * DIM_D + khalf;
      const float* pa = Xs + (r * 16 + mrow) * XS + khalf;
      v8f ag = vz, au = vz;
      // batch 4 k-steps: 12 loads in flight before the first wmma waits
#pragma unroll 1
      for (int kb = 0; kb < DIM_D; kb += 16) {
        v2f A[4], Bg[4], Bu[4];
#pragma unroll
        for (int t = 0; t < 4; ++t) {
          A[t]  = *(const v2f*)(pa + kb + 4 * t);   // ds_load b64, bank-clean
          Bg[t] = *(const v2f*)(pg + kb + 4 * t);   // global b64 (L2 weights)
          Bu[t] = *(const v2f*)(pu + kb + 4 * t);
        }
#pragma unroll
        for (int t = 0; t < 4; ++t) {
          ag = wmma_f32(A[t], Bg[t], ag);
          au = wmma_f32(A[t], Bu[t], au);
        }
      }
      const int rowb = r * 16 + half8;
      const int col  = ct * 16 + mrow;
#pragma unroll
      for (int v = 0; v < 8; ++v) {
        Hs[(rowb + v) * HS + col] = silu_mul(ag[v], au[v]);
      }
    }

    __syncthreads();   // Hs complete for this chunk

    // ---- Phase B: accd += Hs(32 x FC) * Wd[:, c*FC : c*FC+FC]^T ----
#pragma unroll
    for (int i = 0; i < 8; ++i) {
      const int ct2 = wct + 8 * i;            // 0..63 : 16-wide D col tile
      const float* pd = wd + (size_t)(ct2 * 16 + mrow) * DIM_FF + c * FC + khalf;
      const float* ph = Hs + (r * 16 + mrow) * HS + khalf;
      v8f acc = accd[i];
      // batch 8 k-steps: 8 weight loads in flight per wmma group
#pragma unroll 1
      for (int kb = 0; kb < FC; kb += 32) {
        v2f A[8], Bd[8];
#pragma unroll
        for (int t = 0; t < 8; ++t) {
          A[t]  = *(const v2f*)(ph + kb + 4 * t);
          Bd[t] = *(const v2f*)(pd + kb + 4 * t);
        }
#pragma unroll
        for (int t = 0; t < 8; ++t) {
          acc = wmma_f32(A[t], Bd[t], acc);
        }
      }
      accd[i] = acc;
    }
  }

  // ---- Epilogue: scatter 32 x 1024 outputs back to token rows ----
#pragma unroll
  for (int i = 0; i < 8; ++i) {
    const int d0 = (wct + 8 * i) * 16;
#pragma unroll
    for (int v = 0; v < 8; ++v) {
      const int row = r * 16 + half8 + v;
      const int gi = tokIdx[row];
      if (gi >= 0) {
        out[(size_t)gi * DIM_D + d0 + mrow] = accd[i][v];
      }
    }
  }
}

// ---------------------------------------------------------------------------
// Launch: inputs in setup_inputs() order:
//  0 hidden_states f32 (2,2048,1024), 1 token_type_ids int, 2 v_gate (FF,D),
//  3 v_up (FF,D), 4 v_down (D,FF), 5 l_gate, 6 l_up, 7 l_down
// ---------------------------------------------------------------------------
extern "C" void kernel_launch(void* const* d_in, const int* in_sizes, int n_in,
                              void* d_out, int out_size, void* d_ws, size_t ws_size,
                              hipStream_t stream) {
  (void)in_sizes; (void)n_in; (void)out_size; (void)ws_size;
  const float* hs = (const float*)d_in[0];
  const int*   tt = (const int*)d_in[1];      // integer input -> const int*
  const float* vg = (const float*)d_in[2];
  const float* vu = (const float*)d_in[3];
  const float* vd = (const float*)d_in[4];
  const float* lg = (const float*)d_in[5];
  const float* lu = (const float*)d_in[6];
  const float* ld = (const float*)d_in[7];
  float* out = (float*)d_out;

  int* wsI   = (int*)d_ws;
  int* vis   = wsI;                 // capacity N_TOK
  int* lang  = wsI + N_TOK;         // capacity N_TOK
  int* cnts  = wsI + 2 * N_TOK;     // [nvis, nlang]

  mask_compact<<<dim3(1), dim3(1024), 0, stream>>>(tt, vis, lang, cnts);

  const size_t smem = (size_t)(TOK * XS + TOK * HS) * sizeof(float); // ~263 KB of 320 KB
  fused_mlp<<<dim3(N_TOK / TOK, 2), dim3(NTHREADS), smem, stream>>>(
      hs, vg, vu, vd, lg, lu, ld, vis, lang, cnts, out);
}